// KANGraphAttentionLayerARC2_5557687681558
// MI455X (gfx1250) — compile-verified
//
#include <hip/hip_runtime.h>

typedef __attribute__((ext_vector_type(2))) float v2f;
typedef __attribute__((ext_vector_type(8))) float v8f;
typedef __attribute__((ext_vector_type(4))) int   v4i;
typedef __attribute__((address_space(1))) v4i* gv4i_p;  // global int4*
typedef __attribute__((address_space(3))) v4i* lv4i_p;  // LDS int4*

#define FIN   128
#define FOUT  128
#define CB    8          // GRID_SIZE + SPLINE_ORDER
#define KSP   1024       // FIN * CB
#define LALPHA 0.2f

#if __has_builtin(__builtin_amdgcn_global_load_async_to_lds_b128) && \
    __has_builtin(__builtin_amdgcn_s_wait_asynccnt)
#define HAS_ASYNC_LDS 1
#else
#define HAS_ASYNC_LDS 0
#endif

// float atomic max via int punning (works for mixed signs; init with -inf)
__device__ __forceinline__ void atomicMaxFloat(float* addr, float val) {
    if (val >= 0.0f) atomicMax((int*)addr, __float_as_int(val));
    else             atomicMin((unsigned int*)addr, __float_as_uint(val));
}

// ---------------------------------------------------------------------------
// Kernel 1: per 16-node tile, compute HW = h@W and HW_KAN = silu(h)@Bwᵀ +
// bases(h)@Swᵀ with V_WMMA_F32_16X16X4_F32. 8 waves/block, one 16x16 output
// tile per wave. h tile staged via async-LDS copy (if available); silu tile
// precomputed into LDS so the WMMA K-loop is pure loads + wmma; B-spline
// bases staged through LDS in two K=512 halves.
// ---------------------------------------------------------------------------
__global__ __launch_bounds__(256)
void k_node_gemm(const float* __restrict__ h,
                 const float* __restrict__ W,
                 const float* __restrict__ base_w,
                 const float* __restrict__ spline_w,
                 float* __restrict__ HW,
                 float* __restrict__ KAN,
                 int N) {
    __shared__ float h_lds[16][FIN];        // 8 KB
    __shared__ float silu_lds[16][FIN];     // 8 KB
    __shared__ float bases_lds[16][512];    // 32 KB

    const int tid   = threadIdx.x;
    const int wave  = tid >> 5;
    const int lane  = tid & 31;
    const int lh    = lane >> 4;       // lane half: selects K pair per ISA A layout
    const int m     = lane & 15;       // A row / output lane column
    const int n     = wave * 16 + m;   // global output column for this lane
    const int node0 = blockIdx.x * 16;

    // ---- stage h tile into LDS ----
#if HAS_ASYNC_LDS
    // 16 rows x 512B = 512 chunks of 16B; 256 threads -> 2 chunks each.
    for (int c = tid; c < 512; c += 256) {
        int r = c >> 5;            // 32 x 16B chunks per row
        int q = c & 31;
        int gr = node0 + r; if (gr >= N) gr = N - 1;
        const float* gp = h + (size_t)gr * FIN + q * 4;
        float* lp = &h_lds[r][q * 4];
        __builtin_amdgcn_global_load_async_to_lds_b128(
            (gv4i_p)(uintptr_t)gp,
            (lv4i_p)(unsigned)(uintptr_t)lp,
            0, 0);
    }
    __builtin_amdgcn_s_wait_asynccnt(0);
#else
    for (int i = tid; i < 16 * FIN; i += 256) {
        int r = i >> 7, c = i & 127;
        int gr = node0 + r; if (gr >= N) gr = N - 1;
        h_lds[r][c] = h[(size_t)gr * FIN + c];
    }
#endif
    __syncthreads();

    // ---- precompute silu tile (keeps divides out of the WMMA loop) ----
    for (int i = tid; i < 16 * FIN; i += 256) {
        int r = i >> 7, c = i & 127;
        float x = h_lds[r][c];
        silu_lds[r][c] = __fdividef(x, 1.0f + __expf(-x));
    }
    __syncthreads();

    v8f c_hw  = {};
    v8f c_kan = {};

    // HW = h @ W  and  base part: silu(h) @ base_wᵀ   (K = 128)
    for (int k0 = 0; k0 < FIN; k0 += 4) {
        const int kk = k0 + 2 * lh;
        v2f a  = *(const v2f*)&h_lds[m][kk];       // ds_load_b64
        v2f as = *(const v2f*)&silu_lds[m][kk];    // ds_load_b64
        v2f b, bb;
        b.x = W[kk * FOUT + n]; b.y = W[(kk + 1) * FOUT + n];
        bb = *(const v2f*)&base_w[n * FIN + kk];   // global_load_b64
        c_hw  = __builtin_amdgcn_wmma_f32_16x16x4_f32(false, a,  false, b,  (short)0, c_hw,  false, false);
        c_kan = __builtin_amdgcn_wmma_f32_16x16x4_f32(false, as, false, bb, (short)0, c_kan, false, false);
    }

    // spline part: bases @ spline_wᵀ   (K = 1024, staged as two 512 halves)
    for (int ph = 0; ph < 2; ++ph) {
        __syncthreads();
        // cubic B-spline bases for in-features [ph*64, ph*64+64)
        for (int p = tid; p < 16 * 64; p += 256) {
            int r = p >> 6, il = p & 63;
            float x = h_lds[r][ph * 64 + il];
            float bs[11];
            #pragma unroll
            for (int j = 0; j < 11; ++j) {
                float t0 = -1.0f + (float)(j - 3) * 0.4f;
                bs[j] = (x >= t0 && x < t0 + 0.4f) ? 1.0f : 0.0f;
            }
            #pragma unroll
            for (int k = 1; k <= 3; ++k) {
                float inv = 1.0f / (0.4f * (float)k);   // compile-time constant
                #pragma unroll
                for (int j = 0; j + k < 11; ++j) {
                    float tj  = -1.0f + (float)(j - 3) * 0.4f;
                    float tk1 = tj + 0.4f * (float)(k + 1);
                    bs[j] = (x - tj) * inv * bs[j] + (tk1 - x) * inv * bs[j + 1];
                }
            }
            #pragma unroll
            for (int c = 0; c < CB; ++c)
                bases_lds[r][il * CB + c] = bs[c];
        }
        __syncthreads();

        const int kbase = ph * 512;
        const float* swrow = spline_w + (size_t)n * KSP + kbase;
        for (int k0 = 0; k0 < 512; k0 += 4) {
            const int kk = k0 + 2 * lh;
            v2f a = *(const v2f*)&bases_lds[m][kk]; // ds_load_b64
            v2f b = *(const v2f*)&swrow[kk];        // global_load_b64
            c_kan = __builtin_amdgcn_wmma_f32_16x16x4_f32(false, a, false, b, (short)0, c_kan, false, false);
        }
    }

    // store 16x16 tiles per ISA C/D layout: vgpr v -> row v + 8*lh, lane&15 -> col
    #pragma unroll
    for (int v = 0; v < 8; ++v) {
        int r = node0 + v + 8 * lh;
        if (r < N) {
            HW[(size_t)r * FOUT + n]  = c_hw[v];
            KAN[(size_t)r * FOUT + n] = c_kan[v];
        }
    }
}

// ---------------------------------------------------------------------------
// Kernel 2: s_src[i] = HW[i,:]·a[:128], s_dst[i] = HW[i,:]·a[128:]
// wave-per-node, float4 loads + shuffle reduction
// ---------------------------------------------------------------------------
__global__ __launch_bounds__(256)
void k_scores(const float* __restrict__ HW, const float* __restrict__ a,
              float* __restrict__ s_src, float* __restrict__ s_dst, int N) {
    int wave = threadIdx.x >> 5, lane = threadIdx.x & 31;
    int node = blockIdx.x * 8 + wave;
    if (node >= N) return;
    float4 hv = ((const float4*)(HW + (size_t)node * FOUT))[lane];
    float4 av = ((const float4*)a)[lane];
    float4 bv = ((const float4*)a)[32 + lane];
    float ss = hv.x * av.x + hv.y * av.y + hv.z * av.z + hv.w * av.w;
    float sd = hv.x * bv.x + hv.y * bv.y + hv.z * bv.z + hv.w * bv.w;
    for (int off = 16; off > 0; off >>= 1) {
        ss += __shfl_down(ss, off, 32);
        sd += __shfl_down(sd, off, 32);
    }
    if (lane == 0) { s_src[node] = ss; s_dst[node] = sd; }
}

// ---------------------------------------------------------------------------
// Kernel 3: init out=0, m=-inf, denom=0
// ---------------------------------------------------------------------------
__global__ void k_init(float* __restrict__ out, float* __restrict__ m,
                       float* __restrict__ denom, int N) {
    int i = blockIdx.x * blockDim.x + threadIdx.x;
    if (i < N * FOUT) out[i] = 0.0f;
    if (i < N) { m[i] = __uint_as_float(0xFF800000u); denom[i] = 0.0f; }
}

// ---------------------------------------------------------------------------
// Kernel 4: segment max of leaky_relu(s_src[row] + s_dst[col]) by row
// ---------------------------------------------------------------------------
__global__ void k_edge_max(const long long* __restrict__ ei,
                           const float* __restrict__ s_src,
                           const float* __restrict__ s_dst,
                           float* __restrict__ m, int E) {
    int e = blockIdx.x * blockDim.x + threadIdx.x;
    if (e >= E) return;
    int row = (int)ei[e], col = (int)ei[E + e];
    float v = s_src[row] + s_dst[col];
    v = v > 0.0f ? v : LALPHA * v;
    atomicMaxFloat(&m[row], v);
}

// ---------------------------------------------------------------------------
// Kernel 5: e_exp = exp(e - m[row]); denom[row] += e_exp
// ---------------------------------------------------------------------------
__global__ void k_edge_exp(const long long* __restrict__ ei,
                           const float* __restrict__ s_src,
                           const float* __restrict__ s_dst,
                           const float* __restrict__ m,
                           float* __restrict__ eexp,
                           float* __restrict__ denom, int E) {
    int e = blockIdx.x * blockDim.x + threadIdx.x;
    if (e >= E) return;
    int row = (int)ei[e], col = (int)ei[E + e];
    float v = s_src[row] + s_dst[col];
    v = v > 0.0f ? v : LALPHA * v;
    float w = __expf(v - m[row]);
    eexp[e] = w;
    atomicAdd(&denom[row], w);
}

// ---------------------------------------------------------------------------
// Kernel 6: out[row] += (eexp/denom[row]) * HW_KAN[col]; wave per edge,
// float4 gather + fp32 atomic adds (global_atomic_add_f32)
// ---------------------------------------------------------------------------
__global__ void k_scatter(const long long* __restrict__ ei,
                          const float* __restrict__ eexp,
                          const float* __restrict__ denom,
                          const float* __restrict__ KAN,
                          float* __restrict__ out, int E) {
    int gid  = blockIdx.x * blockDim.x + threadIdx.x;
    int e    = gid >> 5;
    int lane = gid & 31;
    if (e >= E) return;
    int row = (int)ei[e], col = (int)ei[E + e];
    float att = __fdividef(eexp[e], denom[row]);
    float4 v = ((const float4*)(KAN + (size_t)col * FOUT))[lane];
    float* dst = out + (size_t)row * FOUT + lane * 4;
    atomicAdd(dst + 0, att * v.x);
    atomicAdd(dst + 1, att * v.y);
    atomicAdd(dst + 2, att * v.z);
    atomicAdd(dst + 3, att * v.w);
}

extern "C" void kernel_launch(void* const* d_in, const int* in_sizes, int n_in,
                              void* d_out, int out_size, void* d_ws, size_t ws_size,
                              hipStream_t stream) {
    const float*     h        = (const float*)d_in[0];
    const long long* ei       = (const long long*)d_in[1]; // int64 [2, E]
    const float*     W        = (const float*)d_in[2];
    const float*     a        = (const float*)d_in[3];
    const float*     base_w   = (const float*)d_in[4];
    const float*     spline_w = (const float*)d_in[5];
    float*           out      = (float*)d_out;

    const int N = in_sizes[0] / FIN;
    const int E = in_sizes[1] / 2;

    float* ws    = (float*)d_ws;
    float* HW    = ws;                         // N*128
    float* KAN   = HW + (size_t)N * FOUT;      // N*128
    float* s_src = KAN + (size_t)N * FOUT;     // N
    float* s_dst = s_src + N;                  // N
    float* m     = s_dst + N;                  // N
    float* denom = m + N;                      // N
    float* eexp  = denom + N;                  // E

    k_node_gemm<<<(N + 15) / 16, 256, 0, stream>>>(h, W, base_w, spline_w, HW, KAN, N);
    k_init<<<(N * FOUT + 255) / 256, 256, 0, stream>>>(out, m, denom, N);
    k_scores<<<(N + 7) / 8, 256, 0, stream>>>(HW, a, s_src, s_dst, N);
    k_edge_max<<<(E + 255) / 256, 256, 0, stream>>>(ei, s_src, s_dst, m, E);
    k_edge_exp<<<(E + 255) / 256, 256, 0, stream>>>(ei, s_src, s_dst, m, eexp, denom, E);
    long long tot = (long long)E * 32;
    k_scatter<<<(int)((tot + 255) / 256), 256, 0, stream>>>(ei, eexp, denom, KAN, out, E);
}